// NormDistBase_54580444397812
// MI455X (gfx1250) — compile-verified
//
#include <hip/hip_runtime.h>
#include <hip/hip_bf16.h>

// Reference problem constants: B=32, Cin=512, Cout=512, L=16, G=1.
#define NB     32
#define CIN    512
#define COUT   512
#define LLEN   16
#define CT     32            // cin tile size
#define COT    64            // couts per block
#define TPB    256           // 8 wave32
#define ROWW   36            // padded LDS row stride in words (bank-conflict free)
#define NTILE  (CIN / CT)    // 16 k-tiles

// ---- CDNA5 async global->LDS path (guarded so we always compile) ----------
#if __has_builtin(__builtin_amdgcn_global_load_async_to_lds_b32) && \
    __has_builtin(__builtin_amdgcn_global_load_async_to_lds_b128)
#define USE_ASYNC 1
#else
#define USE_ASYNC 0
#endif

#if __has_builtin(__builtin_amdgcn_s_wait_asynccnt)
#define WAIT_ASYNCCNT(imm) __builtin_amdgcn_s_wait_asynccnt(imm)
#else
#define WAIT_ASYNCCNT(imm) asm volatile("s_wait_asynccnt " #imm ::: "memory")
#endif

#if USE_ASYNC
#define AS_GLOBAL __attribute__((address_space(1)))
#define AS_LDS    __attribute__((address_space(3)))
typedef int v4i __attribute__((vector_size(4 * sizeof(int))));

__device__ __forceinline__ void cp_async_b32(const float* g, float* l) {
  __builtin_amdgcn_global_load_async_to_lds_b32(
      (AS_GLOBAL int*)(void*)g, (AS_LDS int*)(void*)l, 0, 0);
}
__device__ __forceinline__ void cp_async_b128(const float* g, float* l) {
  __builtin_amdgcn_global_load_async_to_lds_b128(
      (AS_GLOBAL v4i*)(void*)g, (AS_LDS v4i*)(void*)l, 0, 0);
}
#endif

__global__ __launch_bounds__(TPB) void chebydist_ival_kernel(
    const float* __restrict__ x, const float* __restrict__ lo,
    const float* __restrict__ up, const float* __restrict__ W,
    const float* __restrict__ bias, float* __restrict__ out)
{
  // Double-buffered tiles. x/lo/up stored TRANSPOSED as [l][cin_tile].
  __shared__ __align__(16) float sX[2][LLEN * ROWW];
  __shared__ __align__(16) float sL[2][LLEN * ROWW];
  __shared__ __align__(16) float sU[2][LLEN * ROWW];
  __shared__ __align__(16) float sW[2][COT * ROWW];   // [co_local][cin_tile]

  const int t   = threadIdx.x;
  const int l   = t & (LLEN - 1);   // 0..15  : position in L
  const int g   = t >> 4;           // 0..15  : cout group (4 couts each)
  const int b   = blockIdx.y;
  const int co0 = blockIdx.x * COT;

  const float* xb = x  + (size_t)b * CIN * LLEN;
  const float* lb = lo + (size_t)b * CIN * LLEN;
  const float* ub = up + (size_t)b * CIN * LLEN;

  float m [4] = {0.f, 0.f, 0.f, 0.f};  // max |x - W|        (abs >= 0, init 0 ok)
  float ml[4] = {0.f, 0.f, 0.f, 0.f};  // max(max(lo-W, W-up), 0)
  float mu[4] = {0.f, 0.f, 0.f, 0.f};  // max(up-W, W-lo) = |x-W|+eps > 0

  auto compute_tile = [&](int buf) {
    const float4* xr = (const float4*)&sX[buf][l * ROWW];
    const float4* lr = (const float4*)&sL[buf][l * ROWW];
    const float4* ur = (const float4*)&sU[buf][l * ROWW];
#pragma unroll
    for (int cq = 0; cq < CT / 4; ++cq) {
      const float4 xv = xr[cq];
      const float4 lv = lr[cq];
      const float4 uv = ur[cq];
#pragma unroll
      for (int j = 0; j < 4; ++j) {
        const float4 wv = ((const float4*)&sW[buf][(g * 4 + j) * ROWW])[cq];
#define UPD(XE, LE, UE, WE)                           \
        {                                             \
          float d  = (XE) - (WE);                     \
          float a  = (LE) - (WE);                     \
          float bq = (WE) - (UE);                     \
          m [j] = fmaxf(m [j], fabsf(d));             \
          ml[j] = fmaxf(ml[j], fmaxf(a, bq));         \
          mu[j] = fmaxf(mu[j], fmaxf(-a, -bq));       \
        }
        UPD(xv.x, lv.x, uv.x, wv.x)
        UPD(xv.y, lv.y, uv.y, wv.y)
        UPD(xv.z, lv.z, uv.z, wv.z)
        UPD(xv.w, lv.w, uv.w, wv.w)
#undef UPD
      }
    }
  };

#if USE_ASYNC
  // Exactly 8 async instructions per wave per tile => constant-wait double buffer.
  auto issue_tile = [&](int tile, int buf) {
    const int c0 = tile * CT;
    // x/lo/up: b32 per-lane scatter performs the [cin][l] -> [l][cin] transpose.
#pragma unroll
    for (int i = 0; i < 2; ++i) {
      const int w  = i * TPB + t;        // 512 words of the 32x16 tile
      const int cc = w >> 4;
      const int l2 = w & 15;
      const int gs = (c0 + cc) * LLEN + l2;
      const int ld = l2 * ROWW + cc;
      cp_async_b32(xb + gs, &sX[buf][ld]);
      cp_async_b32(lb + gs, &sL[buf][ld]);
      cp_async_b32(ub + gs, &sU[buf][ld]);
    }
    // W: 64 rows x 32 cols, b128 per lane, 2 instructions per wave.
    const int wv  = t >> 5;
    const int lam = t & 31;
#pragma unroll
    for (int j = 0; j < 2; ++j) {
      const int k  = wv * 2 + j;          // 0..15
      const int r  = k * 4 + (lam >> 3);  // 0..63
      const int cc = (lam & 7) * 4;       // 0,4,...,28
      cp_async_b128(W + (size_t)(co0 + r) * CIN + (c0 + cc),
                    &sW[buf][r * ROWW + cc]);
    }
  };

  issue_tile(0, 0);
#pragma unroll 1
  for (int tile = 0; tile < NTILE; ++tile) {
    const int buf = tile & 1;
    if (tile + 1 < NTILE) {
      issue_tile(tile + 1, buf ^ 1);
      WAIT_ASYNCCNT(8);    // tile `tile` landed; next tile may still fly
    } else {
      WAIT_ASYNCCNT(0);
    }
    __syncthreads();       // all waves' tile data visible
    compute_tile(buf);
    __syncthreads();       // reads done before buf is overwritten next iter
  }
#else
  // Synchronous staging fallback (single buffer).
#pragma unroll 1
  for (int tile = 0; tile < NTILE; ++tile) {
    const int c0 = tile * CT;
    __syncthreads();
#pragma unroll
    for (int i = 0; i < 2; ++i) {
      const int w  = i * TPB + t;
      const int cc = w >> 4;
      const int l2 = w & 15;
      const int gs = (c0 + cc) * LLEN + l2;
      const int ld = l2 * ROWW + cc;
      sX[0][ld] = xb[gs];
      sL[0][ld] = lb[gs];
      sU[0][ld] = ub[gs];
    }
    {
      const int wv  = t >> 5;
      const int lam = t & 31;
#pragma unroll
      for (int j = 0; j < 2; ++j) {
        const int k  = wv * 2 + j;
        const int r  = k * 4 + (lam >> 3);
        const int cc = (lam & 7) * 4;
        *(float4*)&sW[0][r * ROWW + cc] =
            *(const float4*)(W + (size_t)(co0 + r) * CIN + (c0 + cc));
      }
    }
    __syncthreads();
    compute_tile(0);
  }
#endif

  // Epilogue: add bias, write the three concatenated output tensors.
  float* out0 = out;
  float* out1 = out0 + (size_t)NB * COUT * LLEN;
  float* out2 = out1 + (size_t)NB * COUT * LLEN;
#pragma unroll
  for (int j = 0; j < 4; ++j) {
    const int co = co0 + g * 4 + j;
    const float bv = bias[co];
    const size_t o = ((size_t)b * COUT + co) * LLEN + l;
    out0[o] = m [j] + bv;
    out1[o] = ml[j] + bv;
    out2[o] = mu[j] + bv;
  }
}

extern "C" void kernel_launch(void* const* d_in, const int* in_sizes, int n_in,
                              void* d_out, int out_size, void* d_ws, size_t ws_size,
                              hipStream_t stream) {
  (void)in_sizes; (void)n_in; (void)d_ws; (void)ws_size; (void)out_size;
  const float* x    = (const float*)d_in[0];
  const float* lo   = (const float*)d_in[1];
  const float* up   = (const float*)d_in[2];
  const float* W    = (const float*)d_in[3];
  const float* bias = (const float*)d_in[4];
  // d_in[5] = groups (==1), hardcoded.
  dim3 grid(COUT / COT, NB);
  chebydist_ival_kernel<<<grid, TPB, 0, stream>>>(x, lo, up, W, bias,
                                                  (float*)d_out);
}